// EdgeUpdateBlock_87608742903967
// MI455X (gfx1250) — compile-verified
//
#include <hip/hip_runtime.h>
#include <math.h>

// ---------------------------------------------------------------------------
// Equivariant EdgeUpdate block for MI455X (gfx1250), wave32 + WMMA f32 16x16x4.
// v2: B matrices pre-packed into WMMA-fragment order (1 coalesced b64 load per
//     WMMA), A tiles stored as even-stride component planes (1 ds b64 per WMMA).
// Memory-bound (~23 GFLOP vs ~350MB @ 23.3 TB/s) -> full fp32 precision.
// ---------------------------------------------------------------------------

typedef float v2f __attribute__((ext_vector_type(2)));
typedef float v8f __attribute__((ext_vector_type(8)));

#define SLD 66
#define VLD 34
#define ELD 132
#define HLD 66
#define VOFF(k) (1056 + (k) * 544)
#define L_XE 0
#define L_NI 2688
#define L_NJ 5376
#define L_EF 8064
#define L_EL 10752
#define L_HB 12928
#define L_WCF 13984
#define L_SHB 15520
#define L_OHB 15904
#define L_AM 15968
#define L_META 15984
#define WAVE_F 16032

#define PCONV(a) ((a) * 13312)
#define PFC0   39936
#define PFC2   48128
#define PFC4   52224
#define PLPRE0 58368
#define PLPRE1 62464
#define PLPOST0 63488
#define PLPOST1 67584
#define PSC0   68608
#define PSC1   84992
#define WP_FLOATS 89088

__device__ __forceinline__ v8f z8() {
  v8f v = {0.f,0.f,0.f,0.f,0.f,0.f,0.f,0.f};
  return v;
}

__device__ __forceinline__ v8f wmma4(v2f a, v2f b, v8f c) {
  return __builtin_amdgcn_wmma_f32_16x16x4_f32(false, a, false, b, (short)0, c,
                                               false, false);
}

// C(16x16) += rowscale(A) @ B.  A: LDS plane (even stride -> ds_load_b64).
// B: packed fragments (lane-contiguous -> one coalesced global_load_b64/WMMA).
__device__ __forceinline__ v8f mm(v8f c, const float* __restrict__ As, int lda,
                                  const float* __restrict__ Bp, int K,
                                  const float* amul, float cmul, int lane) {
  const int half = lane >> 4, mn = lane & 15;
  const float am = (amul ? amul[mn] : 1.0f) * cmul;
  const v2f amv = {am, am};
  const v2f* Ar = (const v2f*)(As + mn * lda + half * 2);
  const v2f* Br = (const v2f*)(Bp + lane * 2);
  for (int k = 0; k < K; k += 4) {
    v2f a = *Ar * amv;                 // v_pk_mul_f32
    c = wmma4(a, *Br, c);
    Ar += 2; Br += 32;
  }
  return c;
}

__device__ __forceinline__ float silu_f(float x) { return x / (1.f + __expf(-x)); }
__device__ __forceinline__ float sigm_f(float x) { return 1.f / (1.f + __expf(-x)); }

// ---------------------------------------------------------------------------
__global__ void k_zero(float* p, int n) {
  int i = blockIdx.x * 256 + threadIdx.x;
  if (i < n) p[i] = 0.f;
}

// ---------------------------------------------------------------------------
// Pack [K][N] row-major weights into WMMA B-fragment order.
struct PackDesc { const float* src; int ld, col0, K, Nt, off; };
struct PackTab { PackDesc d[36]; };

__global__ void k_pack(PackTab T, float* __restrict__ WP) {
  const PackDesc pd = T.d[blockIdx.x];
  const int kk16 = 16 * pd.K;
  const int total = pd.Nt * kk16;
  for (int p = threadIdx.x; p < total; p += 256) {
    int nt = p / kk16;
    int rem = p - nt * kk16;
    int f = rem >> 6;
    int q = rem & 63;
    int lane2 = q >> 1, j = q & 1;
    int half = lane2 >> 4, n = lane2 & 15;
    WP[pd.off + p] = pd.src[(f * 4 + half * 2 + j) * pd.ld + pd.col0 + nt * 16 + n];
  }
}

// ---------------------------------------------------------------------------
__global__ void __launch_bounds__(128)
k_main(const float* __restrict__ node_fea, const float* __restrict__ edge_fea,
       const float* __restrict__ edge_sh, const float* __restrict__ edge_oh,
       const float* __restrict__ edge_len, const int* __restrict__ edge_index,
       const int* __restrict__ batch, const float* __restrict__ WP,
       const float* __restrict__ fcb1, const float* __restrict__ fcb3,
       const float* __restrict__ fcb5,
       float* __restrict__ zpre, float* __restrict__ statsP, int E) {
  extern __shared__ float smem[];
  const int wave = threadIdx.x >> 5;
  const int lane = threadIdx.x & 31;
  const int e0 = (blockIdx.x * 4 + wave) * 16;
  if (e0 >= E) return;                  // wave-uniform (EXEC all-ones for WMMA)

  float* Wm  = smem + wave * WAVE_F;
  float* xe  = Wm + L_XE;
  float* ni  = Wm + L_NI;
  float* nj  = Wm + L_NJ;
  float* ef  = Wm + L_EF;
  float* el  = Wm + L_EL;
  float* hb  = Wm + L_HB;
  float* wcf = Wm + L_WCF;
  float* shb = Wm + L_SHB;
  float* ohb = Wm + L_OHB;
  float* am  = Wm + L_AM;
  int*  meta = (int*)(Wm + L_META);
  float* dbuf = el;
  float* tbuf = el + 544;
  float* gbuf = el;
  float* zg   = ni;

  // ---------------- load phase -------------------------------------------
  if (lane < 16) {
    int e = e0 + lane; if (e >= E) e = E - 1;
    int ii = edge_index[e], jj = edge_index[E + e];
    meta[lane] = ii; meta[16 + lane] = jj; meta[32 + lane] = batch[ii];
    for (int c = 0; c < 9; c++) shb[lane * 24 + c] = edge_sh[e * 9 + c];
    for (int v = 0; v < 4; v++) ohb[lane * 4 + v] = edge_oh[e * 4 + v];
    am[lane] = edge_sh[e * 9 + 0];
    float q0 = edge_sh[e*9+4], q1 = edge_sh[e*9+5], q2 = edge_sh[e*9+6],
          q3 = edge_sh[e*9+7], q4 = edge_sh[e*9+8];
    const float f5 = 0.44721359550f, r2 = 0.70710678119f, r6 = 0.40824829046f;
    float* S = &shb[lane * 24 + 12];
    S[0] = f5*(-q2*r6 + q4*r2); S[1] = f5*q0*r2;            S[2] = f5*q3*r2;
    S[3] = f5*q0*r2;            S[4] = f5*(-q2*r6 - q4*r2); S[5] = f5*q1*r2;
    S[6] = f5*q3*r2;            S[7] = f5*q1*r2;            S[8] = f5*2.f*q2*r6;
  }
  {
    long long base = (long long)e0 * 160;
    long long lim  = (long long)E * 160 - 1;
    for (int t = lane; t < 16 * 160; t += 32) {
      int r = t / 160, c = t - r * 160;
      long long gi = base + t; if (gi > lim) gi = lim;
      float v = edge_fea[gi];
      if (c < 64) xe[r * SLD + c] = v;
      else { int cc = c - 64, u = cc / 3, k = cc - u * 3;
             xe[VOFF(k) + r * VLD + u] = v; }
    }
    long long bl = (long long)e0 * 128;
    long long lml = (long long)E * 128 - 1;
    for (int t = lane; t < 16 * 128; t += 32) {
      int r = t >> 7, c = t & 127;
      long long gi = bl + t; if (gi > lml) gi = lml;
      el[r * ELD + c] = edge_len[gi];
    }
  }
  for (int r = 0; r < 16; r++) {
    int ii = meta[r], jj = meta[16 + r];
    for (int c = lane; c < 160; c += 32) {
      float vi = node_fea[(long long)ii * 160 + c];
      float vj = node_fea[(long long)jj * 160 + c];
      if (c < 64) { ni[r * SLD + c] = vi; nj[r * SLD + c] = vj; }
      else { int cc = c - 64, u = cc / 3, k = cc - u * 3;
             ni[VOFF(k) + r * VLD + u] = vi; nj[VOFF(k) + r * VLD + u] = vj; }
    }
  }
  if (e0 + 64 < E)
    __builtin_prefetch(edge_fea + (long long)(e0 + 64) * 160, 0, 1);

  // ---------------- radial MLP: w = silu(silu(el@W0+b)@W2+b)@W4+b --------
  {
    v8f t4[6];
    const int n = lane & 15, half = lane >> 4;
#pragma unroll
    for (int t = 0; t < 4; t++)
      t4[t] = mm(z8(), el, ELD, WP + PFC0 + t * 2048, 128, nullptr, 1.f, lane);
#pragma unroll
    for (int t = 0; t < 4; t++) {
      int w = t * 16 + n; float bv = fcb1[w];
#pragma unroll
      for (int r = 0; r < 8; r++)
        hb[(half * 8 + r) * HLD + w] = silu_f(t4[t][r] + bv);
    }
#pragma unroll
    for (int t = 0; t < 4; t++)
      t4[t] = mm(z8(), hb, HLD, WP + PFC2 + t * 1024, 64, nullptr, 1.f, lane);
#pragma unroll
    for (int t = 0; t < 4; t++) {
      int w = t * 16 + n; float bv = fcb3[w];
#pragma unroll
      for (int r = 0; r < 8; r++)
        hb[(half * 8 + r) * HLD + w] = silu_f(t4[t][r] + bv);
    }
#pragma unroll
    for (int t = 0; t < 6; t++)
      t4[t] = mm(z8(), hb, HLD, WP + PFC4 + t * 1024, 64, nullptr, 1.f, lane);
#pragma unroll
    for (int t = 0; t < 6; t++) {
      int w = t * 16 + n; float bv = fcb5[w];
#pragma unroll
      for (int r = 0; r < 8; r++)
        wcf[(half * 8 + r) * 96 + w] = t4[t][r] + bv;
    }
  }

  // ---------------- lin_pre ---------------------------------------------
  {
    const int n = lane & 15, half = lane >> 4;
#pragma unroll
    for (int t = 0; t < 4; t++) {
      v8f c = mm(z8(), xe, SLD, WP + PLPRE0 + t * 1024, 64, nullptr, 0.125f, lane);
      int w = t * 16 + n;
#pragma unroll
      for (int r = 0; r < 8; r++) ef[(half * 8 + r) * SLD + w] = c[r];
    }
#pragma unroll
    for (int i = 0; i < 3; i++)
#pragma unroll
      for (int t = 0; t < 2; t++) {
        v8f c = mm(z8(), xe + VOFF(i), VLD, WP + PLPRE1 + t * 512, 32, nullptr,
                   0.17677669529f, lane);
        int w = t * 16 + n;
#pragma unroll
        for (int r = 0; r < 8; r++) ef[VOFF(i) + (half * 8 + r) * VLD + w] = c[r];
      }
  }

  // ---------------- conv_tp (21 paths) ----------------------------------
  v8f C0[4], C1[2], Z2[3][2], TS[2];
#pragma unroll
  for (int t = 0; t < 4; t++) C0[t] = z8();
#pragma unroll
  for (int t = 0; t < 2; t++) { C1[t] = z8(); TS[t] = z8(); }
#pragma unroll
  for (int k = 0; k < 3; k++)
#pragma unroll
    for (int t = 0; t < 2; t++) Z2[k][t] = z8();

  const float* XS[3] = {ni, nj, ef};
  const float s384 = 0.05103103630f;
  const float sTS  = 0.02405626122f;
  const float s864 = 0.03402069087f;
  const float s576 = 1.f / 24.f;
  const float s288 = 0.05892556510f;
#pragma unroll
  for (int a = 0; a < 3; a++) {
    const float* X = XS[a];
    const float* CB = WP + PCONV(a);
#pragma unroll
    for (int t = 0; t < 4; t++)
      C0[t] = mm(C0[t], X, SLD, CB + t * 1024, 64, am, s384, lane);
#pragma unroll
    for (int t = 0; t < 2; t++)
      C1[t] = mm(C1[t], X, SLD, CB + 4096 + t * 1024, 64, am, s384, lane);
#pragma unroll
    for (int t = 0; t < 2; t++)
      TS[t] = mm(TS[t], X, SLD, CB + 6144 + t * 1024, 64, nullptr, sTS, lane);
#pragma unroll
    for (int k = 0; k < 3; k++)
#pragma unroll
      for (int t = 0; t < 2; t++)
        Z2[k][t] = mm(Z2[k][t], X + VOFF(k), VLD, CB + 8192 + t * 512, 32, am,
                      s864, lane);
    for (int idx = lane; idx < 512; idx += 32) {
      int r = idx >> 5, u = idx & 31;
      const float* sr = &shb[r * 24 + 1];
      dbuf[r * VLD + u] = X[VOFF(0) + r * VLD + u] * sr[0]
                        + X[VOFF(1) + r * VLD + u] * sr[1]
                        + X[VOFF(2) + r * VLD + u] * sr[2];
    }
#pragma unroll
    for (int t = 0; t < 4; t++)
      C0[t] = mm(C0[t], dbuf, VLD, CB + 9216 + t * 512, 32, nullptr, s576, lane);
#pragma unroll
    for (int t = 0; t < 2; t++)
      C1[t] = mm(C1[t], dbuf, VLD, CB + 11264 + t * 512, 32, nullptr, s576, lane);
    for (int idx = lane; idx < 512; idx += 32) {
      int r = idx >> 5, u = idx & 31;
      const float* S = &shb[r * 24 + 12];
      float x0 = X[VOFF(0) + r * VLD + u];
      float x1 = X[VOFF(1) + r * VLD + u];
      float x2 = X[VOFF(2) + r * VLD + u];
      tbuf[0 * 544 + r * VLD + u] = x0 * S[0] + x1 * S[3] + x2 * S[6];
      tbuf[1 * 544 + r * VLD + u] = x0 * S[1] + x1 * S[4] + x2 * S[7];
      tbuf[2 * 544 + r * VLD + u] = x0 * S[2] + x1 * S[5] + x2 * S[8];
    }
#pragma unroll
    for (int k = 0; k < 3; k++)
#pragma unroll
      for (int t = 0; t < 2; t++)
        Z2[k][t] = mm(Z2[k][t], tbuf + k * 544, VLD, CB + 12288 + t * 512, 32,
                      nullptr, s288, lane);
  }

  // ------- spill: gate + cfconv folded ----------------------------------
  {
    const int n = lane & 15, half = lane >> 4;
#pragma unroll
    for (int t = 0; t < 2; t++) {
      int w = t * 16 + n;
#pragma unroll
      for (int r = 0; r < 8; r++) gbuf[(half * 8 + r) * 32 + w] = sigm_f(C1[t][r]);
    }
#pragma unroll
    for (int t = 0; t < 4; t++) {
      int w = t * 16 + n;
#pragma unroll
      for (int r = 0; r < 8; r++) {
        int m = half * 8 + r;
        zg[m * SLD + w] = silu_f(C0[t][r]) * wcf[m * 96 + w];
      }
    }
#pragma unroll
    for (int k = 0; k < 3; k++)
#pragma unroll
      for (int t = 0; t < 2; t++) {
        int w = t * 16 + n;
#pragma unroll
        for (int r = 0; r < 8; r++) {
          int m = half * 8 + r;
          float v = Z2[k][t][r] + TS[t][r] * shb[m * 24 + 1 + k];
          zg[VOFF(k) + m * VLD + w] = v * gbuf[m * 32 + w] * wcf[m * 96 + 64 + w];
        }
      }
  }

  // ---------------- lin_post + species shortcut -------------------------
  v8f F0[4], FV[3][2];
#pragma unroll
  for (int t = 0; t < 4; t++) F0[t] = z8();
#pragma unroll
  for (int k = 0; k < 3; k++)
#pragma unroll
    for (int t = 0; t < 2; t++) FV[k][t] = z8();
#pragma unroll
  for (int t = 0; t < 4; t++)
    F0[t] = mm(F0[t], zg, SLD, WP + PLPOST0 + t * 1024, 64, nullptr, 0.125f, lane);
#pragma unroll
  for (int k = 0; k < 3; k++)
#pragma unroll
    for (int t = 0; t < 2; t++)
      FV[k][t] = mm(FV[k][t], zg + VOFF(k), VLD, WP + PLPOST1 + t * 512, 32,
                    nullptr, 0.17677669529f, lane);
  const float scs = 0.0625f;
  const float scv = 0.05103103630f;
#pragma unroll
  for (int v = 0; v < 4; v++) {
    if (lane < 16) am[lane] = ohb[lane * 4 + v];
#pragma unroll
    for (int t = 0; t < 4; t++)
      F0[t] = mm(F0[t], xe, SLD, WP + PSC0 + v * 4096 + t * 1024, 64, am, scs, lane);
#pragma unroll
    for (int k = 0; k < 3; k++)
#pragma unroll
      for (int t = 0; t < 2; t++)
        FV[k][t] = mm(FV[k][t], xe + VOFF(k), VLD, WP + PSC1 + v * 1024 + t * 512,
                      32, am, scv, lane);
  }

  // ---------------- write z_pre + per-graph stats ------------------------
  {
    const int n = lane & 15, half = lane >> 4;
    float ps[8], pq[8], pv[3][8];
#pragma unroll
    for (int r = 0; r < 8; r++) {
      ps[r] = 0.f; pq[r] = 0.f; pv[0][r] = 0.f; pv[1][r] = 0.f; pv[2][r] = 0.f;
    }
    float* zw = zpre + (long long)e0 * 160;
#pragma unroll
    for (int t = 0; t < 4; t++) {
      int w = t * 16 + n;
#pragma unroll
      for (int r = 0; r < 8; r++) {
        int m = half * 8 + r; float v = F0[t][r];
        if (e0 + m < E) zw[m * 160 + w] = v;
        ps[r] += v; pq[r] += v * v;
      }
    }
#pragma unroll
    for (int k = 0; k < 3; k++)
#pragma unroll
      for (int t = 0; t < 2; t++) {
        int w = t * 16 + n;
#pragma unroll
        for (int r = 0; r < 8; r++) {
          int m = half * 8 + r; float v = FV[k][t][r];
          if (e0 + m < E) zw[m * 160 + 64 + w * 3 + k] = v;
          pv[k][r] += v;
        }
      }
    const int slot = blockIdx.x & 63;
#pragma unroll
    for (int r = 0; r < 8; r++) {
      float a0 = ps[r], a1 = pq[r], a2 = pv[0][r], a3 = pv[1][r], a4 = pv[2][r];
#pragma unroll
      for (int mk = 1; mk < 16; mk <<= 1) {
        a0 += __shfl_xor(a0, mk, 32); a1 += __shfl_xor(a1, mk, 32);
        a2 += __shfl_xor(a2, mk, 32); a3 += __shfl_xor(a3, mk, 32);
        a4 += __shfl_xor(a4, mk, 32);
      }
      int m = half * 8 + r;
      if (n == 0 && (e0 + m) < E) {
        float* sp = statsP + ((long long)meta[32 + m] * 64 + slot) * 8;
        atomicAdd(sp + 0, a0); atomicAdd(sp + 1, a1); atomicAdd(sp + 2, a2);
        atomicAdd(sp + 3, a3); atomicAdd(sp + 4, a4); atomicAdd(sp + 5, 1.f);
      }
    }
  }
}

// ---------------------------------------------------------------------------
__global__ void k_reduce(const float* __restrict__ statsP, float* __restrict__ stats) {
  int t = threadIdx.x;
  int g = t >> 3, k = t & 7;
  float s = 0.f;
  for (int slot = 0; slot < 64; slot++) s += statsP[((long long)g * 64 + slot) * 8 + k];
  stats[g * 8 + k] = s;
}

// ---------------------------------------------------------------------------
__global__ void k_final(const float* __restrict__ zpre, const float* __restrict__ stats,
                        const float* __restrict__ edge_fea,
                        const int* __restrict__ edge_index, const int* __restrict__ batch,
                        const float* __restrict__ lnw, const float* __restrict__ lnb,
                        float* __restrict__ out, int E) {
  long long gid = (long long)blockIdx.x * 256 + threadIdx.x;
  if (gid >= (long long)E * 160) return;
  int e = (int)(gid / 160), c = (int)(gid - (long long)e * 160);
  int g = batch[edge_index[e]];
  const float* s = stats + g * 8;
  float deg = fmaxf(s[5], 1.f);
  float v = zpre[gid];
  float val;
  if (c < 64) {
    float mu = s[0] / (64.f * deg);
    float var = s[1] / (64.f * deg) - mu * mu;
    float rn = 1.f / (sqrtf(fmaxf(var, 0.f)) + 1e-5f);
    val = (v - mu) * rn * lnw[c] + lnb[c];
  } else {
    int cc = c - 64, u = cc / 3, d = cc - u * 3;
    float mu = s[2 + d] / (32.f * deg);
    val = (v - mu) * lnw[64 + u];
  }
  out[gid] = edge_fea[gid] + val;
}

// ---------------------------------------------------------------------------
extern "C" void kernel_launch(void* const* d_in, const int* in_sizes, int n_in,
                              void* d_out, int out_size, void* d_ws, size_t ws_size,
                              hipStream_t stream) {
  const float* node_fea   = (const float*)d_in[0];
  const float* edge_fea   = (const float*)d_in[1];
  const float* edge_sh    = (const float*)d_in[2];
  const float* edge_oh    = (const float*)d_in[3];
  const float* edge_len   = (const float*)d_in[4];
  const int*   edge_index = (const int*)d_in[5];
  const int*   batch      = (const int*)d_in[6];
  // params flattened in sorted-key pytree order:
  // conv_tp[21], fc[6], lin_post[2], lin_pre[2], ln_b, ln_w, sc_w[2]
  const float* conv[21];
  for (int i = 0; i < 21; i++) conv[i] = (const float*)d_in[7 + i];
  const float* fc0 = (const float*)d_in[28]; const float* fc1 = (const float*)d_in[29];
  const float* fc2 = (const float*)d_in[30]; const float* fc3 = (const float*)d_in[31];
  const float* fc4 = (const float*)d_in[32]; const float* fc5 = (const float*)d_in[33];
  const float* lpost0 = (const float*)d_in[34]; const float* lpost1 = (const float*)d_in[35];
  const float* lpre0  = (const float*)d_in[36]; const float* lpre1  = (const float*)d_in[37];
  const float* ln_b = (const float*)d_in[38];
  const float* ln_w = (const float*)d_in[39];
  const float* sc0 = (const float*)d_in[40]; const float* sc1 = (const float*)d_in[41];

  const int E = in_sizes[5] / 2;

  float* statsP = (float*)d_ws;            // [16][64][8]
  float* stats  = statsP + 16 * 64 * 8;    // [16][8]
  float* WPd    = stats + 128;             // packed weights [89088]
  float* zpre   = WPd + WP_FLOATS;         // [E][160]

  const int nz = 16 * 64 * 8 + 128;
  k_zero<<<(nz + 255) / 256, 256, 0, stream>>>(statsP, nz);

  // ---- build pack table ----
  PackTab T;
  int di = 0;
  for (int a = 0; a < 3; a++) {
    const int cb = a * 7, po = PCONV(a);
    T.d[di++] = {conv[cb + 0], 64, 0, 64, 4, po + 0};
    T.d[di++] = {conv[cb + 1], 32, 0, 64, 2, po + 4096};
    T.d[di++] = {conv[cb + 2], 32, 0, 64, 2, po + 6144};
    T.d[di++] = {conv[cb + 3], 32, 0, 32, 2, po + 8192};
    T.d[di++] = {conv[cb + 4], 64, 0, 32, 4, po + 9216};
    T.d[di++] = {conv[cb + 5], 32, 0, 32, 2, po + 11264};
    T.d[di++] = {conv[cb + 6], 32, 0, 32, 2, po + 12288};
  }
  T.d[di++] = {fc0, 64, 0, 128, 4, PFC0};
  T.d[di++] = {fc2, 64, 0, 64, 4, PFC2};
  T.d[di++] = {fc4, 96, 0, 64, 6, PFC4};
  T.d[di++] = {lpre0, 64, 0, 64, 4, PLPRE0};
  T.d[di++] = {lpre1, 32, 0, 32, 2, PLPRE1};
  T.d[di++] = {lpost0, 64, 0, 64, 4, PLPOST0};
  T.d[di++] = {lpost1, 32, 0, 32, 2, PLPOST1};
  for (int v = 0; v < 4; v++)
    T.d[di++] = {sc0, 256, v * 64, 64, 4, PSC0 + v * 4096};
  for (int v = 0; v < 4; v++)
    T.d[di++] = {sc1, 128, v * 32, 32, 2, PSC1 + v * 1024};
  k_pack<<<36, 256, 0, stream>>>(T, WPd);

  const int tiles = (E + 15) / 16;
  const int blocks = (tiles + 3) / 4;
  const size_t shmem = (size_t)4 * WAVE_F * sizeof(float);  // ~250KB of 320KB LDS
  k_main<<<blocks, 128, shmem, stream>>>(node_fea, edge_fea, edge_sh, edge_oh,
                                         edge_len, edge_index, batch, WPd,
                                         fc1, fc3, fc5, zpre, statsP, E);

  k_reduce<<<1, 128, 0, stream>>>(statsP, stats);

  long long tot = (long long)E * 160;
  k_final<<<(int)((tot + 255) / 256), 256, 0, stream>>>(zpre, stats, edge_fea,
                                                        edge_index, batch, ln_w, ln_b,
                                                        (float*)d_out, E);
}